// CopyDecoder_5274219840242
// MI455X (gfx1250) — compile-verified
//
#include <hip/hip_runtime.h>
#include <hip/hip_bf16.h>

typedef __attribute__((ext_vector_type(2))) float v2f;
typedef __attribute__((ext_vector_type(8))) float v8f;
typedef __attribute__((ext_vector_type(4))) int   v4i;

#define ASG __attribute__((address_space(1)))
#define ASL __attribute__((address_space(3)))

#if defined(__has_builtin)
#if __has_builtin(__builtin_amdgcn_global_load_async_to_lds_b128)
#define USE_ASYNC_LDS 1
#endif
#endif

#define D_MODEL 512
#define NUM_HEADS 8
#define DH 64
#define VOCAB 32000
#define TS 512

// ---------------------------------------------------------------------------
// GEMM: C[M,N] = A[M,K] @ W[N,K]^T + bias[N]   (all fp32, row-major)
// Block = 256 threads = 8 waves; block tile 32(M) x 64(N); waves 2x4, each
// wave owns a 16x16 C tile computed with V_WMMA_F32_16X16X4_F32.
// Tile staging: 128-bit async-to-LDS (ASYNCcnt) when available, float4 else.
// ---------------------------------------------------------------------------
#define TILE_M 32
#define TILE_N 64
#define TILE_K 32

__global__ __launch_bounds__(256) void wmma_gemm_bias(
    const float* __restrict__ A, const float* __restrict__ W,
    const float* __restrict__ bias, float* __restrict__ C,
    int M, int N, int K)
{
    __shared__ float As[TILE_M][TILE_K];   // 4 KB
    __shared__ float Ws[TILE_N][TILE_K];   // 8 KB

    const int tid  = threadIdx.x;
    const int wave = tid >> 5;           // 0..7
    const int lane = tid & 31;
    const int wm   = wave >> 2;          // 0..1 : M direction
    const int wn   = wave & 3;           // 0..3 : N direction
    const int bm   = blockIdx.x * TILE_M;
    const int bn   = blockIdx.y * TILE_N;
    const int l15  = lane & 15;
    const int lhi  = lane >> 4;          // 0 or 1

    // 16-byte staging chunks: A tile = 256 chunks (1/thread), W = 512 (2/thread)
    const int ra  = tid >> 3;            // 0..31
    const int ca  = (tid & 7) * 4;       // 0..28
    const int rw1 = ra + 32;             // 32..63

    v8f acc = {};

    for (int k0 = 0; k0 < K; k0 += TILE_K) {
#ifdef USE_ASYNC_LDS
        __builtin_amdgcn_global_load_async_to_lds_b128(
            (ASG v4i*)&A[(size_t)(bm + ra) * K + k0 + ca],
            (ASL v4i*)&As[ra][ca], 0, 0);
        __builtin_amdgcn_global_load_async_to_lds_b128(
            (ASG v4i*)&W[(size_t)(bn + ra) * K + k0 + ca],
            (ASL v4i*)&Ws[ra][ca], 0, 0);
        __builtin_amdgcn_global_load_async_to_lds_b128(
            (ASG v4i*)&W[(size_t)(bn + rw1) * K + k0 + ca],
            (ASL v4i*)&Ws[rw1][ca], 0, 0);
        asm volatile("s_wait_asynccnt 0x0" ::: "memory");
#else
        *(float4*)&As[ra][ca]  = *(const float4*)&A[(size_t)(bm + ra) * K + k0 + ca];
        *(float4*)&Ws[ra][ca]  = *(const float4*)&W[(size_t)(bn + ra) * K + k0 + ca];
        *(float4*)&Ws[rw1][ca] = *(const float4*)&W[(size_t)(bn + rw1) * K + k0 + ca];
#endif
        __syncthreads();

        #pragma unroll
        for (int kk = 0; kk < TILE_K; kk += 4) {
            const int ka = kk + lhi * 2;                 // this lane's K pair
            v2f a = *(const v2f*)&As[wm * 16 + l15][ka];
            v2f b = *(const v2f*)&Ws[wn * 16 + l15][ka]; // B[k][n] = W[n][k]
            acc = __builtin_amdgcn_wmma_f32_16x16x4_f32(
                false, a, false, b, (short)0, acc, false, false);
        }
        __syncthreads();
    }

    const int n  = bn + wn * 16 + l15;
    const float bv = bias[n];
    #pragma unroll
    for (int r = 0; r < 8; ++r) {
        int m = bm + wm * 16 + r + lhi * 8;      // C layout: lanes>=16 hold M+8
        C[(size_t)m * N + n] = acc[r] + bv;
    }
}

// ---------------------------------------------------------------------------
// Attention: per (b, 16-query tile) block; loop heads, WMMA scores (Q.K^T),
// softmax over ts=512 in LDS, head-averaged attn accumulated in LDS.
// Q, Kp: [B*512, 512] fp32 (head h occupies columns h*64 .. h*64+63).
// ---------------------------------------------------------------------------
__global__ __launch_bounds__(256) void attn_kernel(
    const float* __restrict__ Q, const float* __restrict__ Kp,
    float* __restrict__ attn)
{
    extern __shared__ float smem[];
    float* Qs     = smem;               // 16*64   = 1024 floats
    float* scores = Qs + 16 * DH;       // 16*512  = 8192 floats
    float* accum  = scores + 16 * TS;   // 16*512  = 8192 floats
    float* red    = accum + 16 * TS;    // 512 floats (two reduce buffers)

    const int blk  = blockIdx.x;        // b*32 + qt
    const int b    = blk >> 5;
    const int q0   = (blk & 31) * 16;
    const int tid  = threadIdx.x;
    const int wave = tid >> 5;
    const int lane = tid & 31;
    const int l15  = lane & 15;
    const int lhi  = lane >> 4;
    const int rowbase = b * 512 + q0;

    float4* acc4 = (float4*)accum;
    for (int i = tid; i < (16 * TS) / 4; i += 256)
        acc4[i] = make_float4(0.f, 0.f, 0.f, 0.f);

    for (int h = 0; h < NUM_HEADS; ++h) {
        __syncthreads();
        // stage this head's 16x64 Q tile (one float4 per thread)
        {
            int r = tid >> 4, c4 = (tid & 15) * 4;
            *(float4*)&Qs[r * DH + c4] =
                *(const float4*)&Q[(size_t)(rowbase + r) * D_MODEL + h * DH + c4];
        }
        __syncthreads();

        // each wave computes 4 adjacent 16x16 score tiles (64 key cols)
        #pragma unroll
        for (int t = 0; t < 4; ++t) {
            const int s0 = wave * 64 + t * 16;
            v8f acc = {};
            #pragma unroll
            for (int kk = 0; kk < DH; kk += 4) {
                const int ka = kk + lhi * 2;
                v2f a = *(const v2f*)&Qs[l15 * DH + ka];
                v2f bb = *(const v2f*)
                    &Kp[(size_t)(b * 512 + s0 + l15) * D_MODEL + h * DH + ka];
                acc = __builtin_amdgcn_wmma_f32_16x16x4_f32(
                    false, a, false, bb, (short)0, acc, false, false);
            }
            #pragma unroll
            for (int r = 0; r < 8; ++r) {
                int m = r + lhi * 8;
                scores[m * TS + s0 + l15] = acc[r] * 0.125f;  // 1/sqrt(64)
            }
        }
        __syncthreads();

        // softmax over 512 keys per row; 16 lanes/row, each owns 32 contig cols
        const int row = tid >> 4;
        const int lr  = tid & 15;
        float4* srow = (float4*)&scores[row * TS];
        float4* arow = (float4*)&accum[row * TS];

        float mx = -1e30f;
        #pragma unroll
        for (int c = 0; c < 8; ++c) {
            float4 tv = srow[lr * 8 + c];
            mx = fmaxf(mx, fmaxf(fmaxf(tv.x, tv.y), fmaxf(tv.z, tv.w)));
        }
        red[row * 16 + lr] = mx;
        __syncthreads();
        if (lr == 0) {
            float m2 = red[row * 16];
            for (int j = 1; j < 16; ++j) m2 = fmaxf(m2, red[row * 16 + j]);
            red[row * 16] = m2;
        }
        __syncthreads();
        const float m2 = red[row * 16];
        float sm = 0.0f;
        #pragma unroll
        for (int c = 0; c < 8; ++c) {
            float4 tv = srow[lr * 8 + c];
            tv.x = __expf(tv.x - m2); tv.y = __expf(tv.y - m2);
            tv.z = __expf(tv.z - m2); tv.w = __expf(tv.w - m2);
            srow[lr * 8 + c] = tv;                // keep exp for accumulate pass
            sm += tv.x + tv.y + tv.z + tv.w;
        }
        red[256 + row * 16 + lr] = sm;
        __syncthreads();
        if (lr == 0) {
            float s2 = 0.0f;
            for (int j = 0; j < 16; ++j) s2 += red[256 + row * 16 + j];
            red[256 + row * 16] = s2;
        }
        __syncthreads();
        const float inv = 0.125f / red[256 + row * 16];   // includes 1/NUM_HEADS
        #pragma unroll
        for (int c = 0; c < 8; ++c) {
            float4 e = srow[lr * 8 + c];
            float4 u = arow[lr * 8 + c];
            u.x += e.x * inv; u.y += e.y * inv;
            u.z += e.z * inv; u.w += e.w * inv;
            arow[lr * 8 + c] = u;
        }
    }
    __syncthreads();

    for (int i = tid; i < (16 * TS) / 4; i += 256) {
        int r = i >> 7, c4 = (i & 127) * 4;
        *(float4*)&attn[(size_t)(rowbase + r) * TS + c4] =
            *(const float4*)&accum[r * TS + c4];
    }
}

// ---------------------------------------------------------------------------
// Fused copy-distribution: one block per (b,q) row. The implicit a_output
// row [32000] lives entirely in LDS (128 KB of CDNA5's 320KB WGP LDS):
// scatter-add of 512 attn weights by src id, softmax over vocab, fused
// sigmoid gate + blend with p1. All vocab passes are float4 (b128).
// ---------------------------------------------------------------------------
__global__ __launch_bounds__(256) void copy_dist_kernel(
    const float* __restrict__ attn, const int* __restrict__ src,
    const float* __restrict__ p1, const float* __restrict__ dec,
    const float* __restrict__ Wfcw, const float* __restrict__ bfcw,
    float* __restrict__ out)
{
    extern __shared__ float a[];                 // VOCAB floats = 128 KB
    __shared__ float red[256];
    __shared__ float sh_w, sh_m, sh_d;

    const int row = blockIdx.x;                  // b*512 + q
    const int b   = row >> 9;
    const int tid = threadIdx.x;
    const int V4  = VOCAB / 4;                   // 8000

    float4* a4 = (float4*)a;
    for (int v = tid; v < V4; v += 256) a4[v] = make_float4(0.f, 0.f, 0.f, 0.f);

    // warm L2->WGP path for the p1 row we'll stream in the final pass
    const char* p1c = (const char*)(p1 + (size_t)row * VOCAB);
    for (int i = tid; i < (VOCAB * 4) / 128; i += 256)
        __builtin_prefetch(p1c + i * 128, 0, 1);

    // gate: w = sigmoid(dec_row . Wfcw + bfcw)
    float pd = 0.0f;
    for (int k = tid; k < D_MODEL; k += 256)
        pd += dec[(size_t)row * D_MODEL + k] * Wfcw[k];
    red[tid] = pd;
    __syncthreads();
    for (int s = 128; s > 0; s >>= 1) {
        if (tid < s) red[tid] += red[tid + s];
        __syncthreads();
    }
    if (tid == 0) {
        float z = red[0] + bfcw[0];
        sh_w = 1.0f / (1.0f + __expf(-z));
    }

    // scatter-add attention weights into LDS vocab buffer
    for (int s = tid; s < TS; s += 256)
        atomicAdd(&a[src[b * TS + s]], attn[(size_t)row * TS + s]);
    __syncthreads();

    // max over vocab (zero baseline included automatically)
    float mx = 0.0f;
    for (int v = tid; v < V4; v += 256) {
        float4 t = a4[v];
        mx = fmaxf(mx, fmaxf(fmaxf(t.x, t.y), fmaxf(t.z, t.w)));
    }
    red[tid] = mx;
    __syncthreads();
    for (int s = 128; s > 0; s >>= 1) {
        if (tid < s) red[tid] = fmaxf(red[tid], red[tid + s]);
        __syncthreads();
    }
    if (tid == 0) sh_m = red[0];
    __syncthreads();

    // exp + sum; store exp back into LDS
    const float m = sh_m;
    float sm = 0.0f;
    for (int v = tid; v < V4; v += 256) {
        float4 t = a4[v];
        t.x = __expf(t.x - m); t.y = __expf(t.y - m);
        t.z = __expf(t.z - m); t.w = __expf(t.w - m);
        a4[v] = t;
        sm += t.x + t.y + t.z + t.w;
    }
    red[tid] = sm;
    __syncthreads();
    for (int s = 128; s > 0; s >>= 1) {
        if (tid < s) red[tid] += red[tid + s];
        __syncthreads();
    }
    if (tid == 0) sh_d = red[0];
    __syncthreads();

    // fused blend: out = p2 * w + (1-w) * p1   (single pass, b128 traffic)
    const float w     = sh_w;
    const float scale = w / sh_d;
    const float om    = 1.0f - w;
    const float4* p14 = (const float4*)(p1 + (size_t)row * VOCAB);
    float4* out4      = (float4*)(out + (size_t)row * VOCAB);
    for (int v = tid; v < V4; v += 256) {
        float4 e = a4[v];
        float4 p = p14[v];
        float4 o;
        o.x = e.x * scale + om * p.x;
        o.y = e.y * scale + om * p.y;
        o.z = e.z * scale + om * p.z;
        o.w = e.w * scale + om * p.w;
        out4[v] = o;
    }
}

// ---------------------------------------------------------------------------
extern "C" void kernel_launch(void* const* d_in, const int* in_sizes, int n_in,
                              void* d_out, int out_size, void* d_ws, size_t ws_size,
                              hipStream_t stream) {
    const float* dec  = (const float*)d_in[1];
    const float* enc  = (const float*)d_in[2];
    const int*   src  = (const int*)d_in[3];
    const float* p1   = (const float*)d_in[4];
    const float* WfcQ = (const float*)d_in[5];
    const float* bfcQ = (const float*)d_in[6];
    const float* Wq   = (const float*)d_in[7];
    const float* bq   = (const float*)d_in[8];
    const float* Wk   = (const float*)d_in[9];
    const float* bk   = (const float*)d_in[10];
    const float* Wfcw = (const float*)d_in[11];
    const float* bfcw = (const float*)d_in[12];
    float* out = (float*)d_out;

    const int M = 4 * 512;                     // B * TQ (= B * TS)
    const int N = D_MODEL, K = D_MODEL;

    char* ws = (char*)d_ws;
    const size_t mat = (size_t)M * D_MODEL * sizeof(float);   // 4 MB each
    float* cq   = (float*)(ws);
    float* Qp   = (float*)(ws + mat);
    float* Kp   = (float*)(ws + 2 * mat);
    float* attn = (float*)(ws + 3 * mat);

    dim3 gg(M / TILE_M, N / TILE_N);
    wmma_gemm_bias<<<gg, 256, 0, stream>>>(dec, WfcQ, bfcQ, cq, M, N, K);
    wmma_gemm_bias<<<gg, 256, 0, stream>>>(cq,  Wq,   bq,   Qp, M, N, K);
    wmma_gemm_bias<<<gg, 256, 0, stream>>>(enc, Wk,   bk,   Kp, M, N, K);

    const size_t attn_smem =
        (16 * DH + 16 * TS + 16 * TS + 512) * sizeof(float); // ~70 KB
    attn_kernel<<<4 * 32, 256, attn_smem, stream>>>(Qp, Kp, attn);

    copy_dist_kernel<<<M, 256, (size_t)VOCAB * sizeof(float), stream>>>(
        attn, src, p1, dec, Wfcw, bfcw, out);
}